// CGNN_41137196761319
// MI455X (gfx1250) — compile-verified
//
#include <hip/hip_runtime.h>

// MI455X / gfx1250, wave32. fp32 WMMA (V_WMMA_F32_16X16X4_F32) everywhere.
// K1: LDS-resident GNN with transposed double-buffered W panels (tight
//     ds_load_2addr_b64 + wmma loop).
// K2: HBM-streaming classifier GEMM with async-to-LDS double buffering
//     (GLOBAL_LOAD_ASYNC_TO_LDS_B128 + s_wait_asynccnt).
// d_ws requirement: 2048*32768*4 = 256 MiB for the h_flat spill.

typedef float v2f __attribute__((ext_vector_type(2)));
typedef float v8f __attribute__((ext_vector_type(8)));
typedef int   v4i __attribute__((vector_size(4 * sizeof(int))));

#define BATCH   2048
#define NFEAT   128     // graph nodes per batch row
#define HID     256
#define NLAYERS 3
#define NEDGES  4096
#define OUTD    2

// ---------------- async-to-LDS plumbing (probe via __has_builtin) ----------
#if defined(__has_builtin)
#  if __has_builtin(__builtin_amdgcn_global_load_async_to_lds_b128)
#    define HAVE_ASYNC_LDS 1
#  endif
#endif
#ifndef HAVE_ASYNC_LDS
#  define HAVE_ASYNC_LDS 0
#endif

#if HAVE_ASYNC_LDS
#  if __has_builtin(__builtin_amdgcn_s_wait_asynccnt)
#    define WAIT_ASYNC() __builtin_amdgcn_s_wait_asynccnt(0)
#  else
#    define WAIT_ASYNC() asm volatile("s_wait_asynccnt 0" ::: "memory")
#  endif
#else
#  define WAIT_ASYNC() ((void)0)
#endif

// copy 16 contiguous bytes global -> LDS (async DMA if available)
__device__ __forceinline__ void cp16(const float* __restrict__ g, float* l)
{
#if HAVE_ASYNC_LDS
    __builtin_amdgcn_global_load_async_to_lds_b128(
        (__attribute__((address_space(1))) v4i*)(v4i*)g,   // drop const, AS1
        (__attribute__((address_space(3))) v4i*)(v4i*)l,   // generic -> LDS
        /*offset=*/0, /*cpol=*/0);
#else
    *(float4*)l = *(const float4*)g;
#endif
}

// -------------------------------------------------------------------------
// Kernel 1: fused encoder + 3 GNN layers. One batch per workgroup.
// LDS-resident h/h_new; W staged per 16-col N-panel (transposed layout so
// B-fragments are contiguous ds_load_b64), double buffered.
// -------------------------------------------------------------------------
#define H_STRIDE  260   // bank = (4*row + k) % 64: conflict-free fragment reads
#define WP_STRIDE 260

__global__ __launch_bounds__(256, 1)
void gnn_fused(const float* __restrict__ x,    const float* __restrict__ ea,
               const float* __restrict__ encw, const float* __restrict__ encb,
               const float* __restrict__ lw,   const float* __restrict__ lb,
               const int*   __restrict__ eidx, float* __restrict__ hout)
{
    __shared__ float hS [NFEAT][H_STRIDE];    // 133,120 B  current node state
    __shared__ float hnS[NFEAT][H_STRIDE];    // 133,120 B  h_new (post-GEMM ReLU)
    __shared__ float wpT[2][16][WP_STRIDE];   //  33,280 B  W N-panels, transposed

    const int b    = blockIdx.x;
    const int tid  = threadIdx.x;
    const int wave = tid >> 5;       // 8 waves
    const int lane = tid & 31;

    // ---- encoder: h[n][k] = x[b][n]*enc_w[k] + enc_b[k]
    for (int idx = tid; idx < NFEAT * HID; idx += 256) {
        const int n = idx >> 8, k = idx & 255;
        hS[n][k] = x[b * NFEAT + n] * encw[k] + encb[k];
    }

    // fragment lane mapping (16x4 f32): lanes 0-15 -> K0/K1, 16-31 -> K2/K3
    const int mrow  = lane & 15;
    const int khalf = (lane >> 4) << 1;      // 0 or 2
    const int rbase = ((lane >> 4) << 3);    // C/D: VGPR i -> M = i (+8 hi half)

    for (int l = 0; l < NLAYERS; ++l) {
        const float* W  = lw + l * HID * HID;   // W[h][k]: row = K-dim, col = N
        const float* Bv = lb + l * HID;

        // cooperative transposed staging of W[:, nt*16..+15] into wpT[buf]
        auto stage_w = [&](int buf, int nt) {
            for (int i = 0; i < 16; ++i) {        // 4096 elems, 16 per thread
                const int idx = i * 256 + tid;
                const int r = idx >> 4, c = idx & 15;
                wpT[buf][c][r] = W[r * HID + nt * 16 + c];
            }
        };

        stage_w(0, 0);
        __syncthreads();                          // also covers encoder / prev layer

        for (int nt = 0; nt < 16; ++nt) {
            const int cur = nt & 1;
            if (nt < 15) stage_w(cur ^ 1, nt + 1);   // prefetch next N-panel

            // each wave: one 16x16 tile, K = 256 in steps of 4
            v8f acc = {};
            const int m = wave * 16 + mrow;
            for (int k = 0; k < HID; k += 4) {
                v2f a, bb;
                a.x  = hS[m][k + khalf];                  // ds_load_b64
                a.y  = hS[m][k + khalf + 1];
                bb.x = wpT[cur][mrow][k + khalf];         // ds_load_b64
                bb.y = wpT[cur][mrow][k + khalf + 1];
                acc = __builtin_amdgcn_wmma_f32_16x16x4_f32(
                          false, a, false, bb, (short)0, acc, false, false);
            }
            // epilogue: bias + relu -> h_new
            const int   ncol = nt * 16 + mrow;
            const float bias = Bv[ncol];
            #pragma unroll
            for (int i = 0; i < 8; ++i)
                hnS[wave * 16 + rbase + i][ncol] = fmaxf(acc[i] + bias, 0.f);
            __syncthreads();
        }

        // ---- message pass: h += sum_e w_e * h_new[src_e] at dst_e (ds_add_f32)
        for (int e = tid; e < NEDGES; e += 256) {
            const int   s = eidx[e];
            const int   d = eidx[NEDGES + e];
            const float w = ea[e];
            for (int k = 0; k < HID; ++k)
                atomicAdd(&hS[d][k], w * hnS[s][k]);
        }
        __syncthreads();

        // ---- residual ReLU in place: h = relu(h_old + messages)
        for (int idx = tid; idx < NFEAT * HID; idx += 256) {
            const int n = idx >> 8, k = idx & 255;
            hS[n][k] = fmaxf(hS[n][k], 0.f);
        }
        __syncthreads();
    }

    // ---- spill h_flat (row-major b x 32768) for the classifier GEMM
    for (int idx = tid; idx < NFEAT * HID; idx += 256) {
        const int n = idx >> 8, k = idx & 255;
        hout[(size_t)b * (NFEAT * HID) + idx] = hS[n][k];
    }
}

// -------------------------------------------------------------------------
// Kernel 2: classifier GEMM 2048 x 32768 x 256 (+bias+ReLU) fused with the
// 256 -> 2 head. Block tile 32(M) x 256(N); double-buffered async K-panels.
// -------------------------------------------------------------------------
#define MT         32
#define KB         64
#define SA_STRIDE  68   // 272B rows: 16B aligned b128 dst, (4m+k)%64 banks
#define SB_STRIDE  260  // 1040B rows: 16B aligned, padded banks
#define SH_STRIDE  260

__global__ __launch_bounds__(256, 1)
void classifier(const float* __restrict__ hflat, const float* __restrict__ w1,
                const float* __restrict__ b1,    const float* __restrict__ w2,
                const float* __restrict__ b2,    float* __restrict__ out)
{
    __shared__ float sA[2][MT][SA_STRIDE];   //  17,408 B
    __shared__ float sB[2][KB][SB_STRIDE];   // 133,120 B
    __shared__ float sH[MT][SH_STRIDE];      //  33,280 B

    const int tid  = threadIdx.x;
    const int wave = tid >> 5;
    const int lane = tid & 31;
    const int row0 = blockIdx.x * MT;
    const int KTOT = NFEAT * HID;            // 32768

    const int mt    = wave & 1;              // 2 M-tiles
    const int ntb   = (wave >> 1) * 4;       // 4 N-tiles per wave (16 total)
    const int mrow  = lane & 15;
    const int khalf = (lane >> 4) << 1;
    const int rbase = ((lane >> 4) << 3);

    v8f acc[4] = {{}, {}, {}, {}};

    // stage(panel buf, k-offset kb): A 32x64 (HBM), B 64x256 (L2-resident w1)
    auto stage = [&](int buf, int kb) {
        for (int i = 0; i < 2; ++i) {              // A: 512 async b128 lane-ops
            const int idx = i * 256 + tid;         // 0..511
            const int r = idx >> 4, c4 = (idx & 15) << 2;
            cp16(hflat + (size_t)(row0 + r) * KTOT + kb + c4, &sA[buf][r][c4]);
        }
        for (int i = 0; i < 16; ++i) {             // B: 4096 async b128 lane-ops
            const int idx = i * 256 + tid;         // 0..4095
            const int r = idx >> 6, c4 = (idx & 63) << 2;
            cp16(w1 + (size_t)(kb + r) * HID + c4, &sB[buf][r][c4]);
        }
    };

    stage(0, 0);
    WAIT_ASYNC();
    __syncthreads();

    int p = 0;
    for (int kb = 0; kb < KTOT; kb += KB, p ^= 1) {
        if (kb + KB < KTOT) stage(p ^ 1, kb + KB);   // async prefetch next panel

        const int m = mt * 16 + mrow;
        for (int kk = 0; kk < KB; kk += 4) {
            v2f a;
            a.x = sA[p][m][kk + khalf];
            a.y = sA[p][m][kk + khalf + 1];
            #pragma unroll
            for (int j = 0; j < 4; ++j) {
                v2f bb;
                const int ncol = (ntb + j) * 16 + mrow;
                bb.x = sB[p][kk + khalf][ncol];
                bb.y = sB[p][kk + khalf + 1][ncol];
                acc[j] = __builtin_amdgcn_wmma_f32_16x16x4_f32(
                             false, a, false, bb, (short)0, acc[j], false, false);
            }
        }
        WAIT_ASYNC();                                 // drain prefetch
        __syncthreads();
    }

    // epilogue: bias + relu -> hidden tile in LDS
    const int rb = mt * 16 + rbase;
    #pragma unroll
    for (int j = 0; j < 4; ++j) {
        const int   ncol = (ntb + j) * 16 + mrow;
        const float bias = b1[ncol];
        #pragma unroll
        for (int i = 0; i < 8; ++i)
            sH[rb + i][ncol] = fmaxf(acc[j][i] + bias, 0.f);
    }
    __syncthreads();

    // fused head: logits[32][2] = hidden @ w2 + b2
    if (tid < MT * OUTD) {
        const int r = tid >> 1, c = tid & 1;
        float s = 0.f;
        for (int k = 0; k < HID; ++k)
            s += sH[r][k] * w2[k * OUTD + c];
        out[(size_t)(row0 + r) * OUTD + c] = s + b2[c];
    }
}

// -------------------------------------------------------------------------
extern "C" void kernel_launch(void* const* d_in, const int* in_sizes, int n_in,
                              void* d_out, int out_size, void* d_ws, size_t ws_size,
                              hipStream_t stream)
{
    const float* x    = (const float*)d_in[0];
    const float* ea   = (const float*)d_in[1];
    const float* encw = (const float*)d_in[2];
    const float* encb = (const float*)d_in[3];
    const float* lw   = (const float*)d_in[4];
    const float* lb   = (const float*)d_in[5];
    const float* w1   = (const float*)d_in[6];
    const float* b1   = (const float*)d_in[7];
    const float* w2   = (const float*)d_in[8];
    const float* b2   = (const float*)d_in[9];
    const int*   ei   = (const int*)d_in[10];
    float* out   = (float*)d_out;
    float* hflat = (float*)d_ws;   // 2048 * 32768 floats = 256 MiB scratch

    gnn_fused<<<BATCH, 256, 0, stream>>>(x, ea, encw, encb, lw, lb, ei, hflat);
    classifier<<<BATCH / MT, 256, 0, stream>>>(hflat, w1, b1, w2, b2, out);
}